// _multiCodebookQuantization_49701361550144
// MI455X (gfx1250) — compile-verified
//
#include <hip/hip_runtime.h>

typedef __attribute__((ext_vector_type(2))) float v2f;
typedef __attribute__((ext_vector_type(4))) float v4f;
typedef __attribute__((ext_vector_type(8))) float v8f;

#define N_ 4096
#define M_ 4
#define K_ 2048
#define D_ 32
#define TN 16
#define NWAVE 8
#define KT_PER_WAVE ((K_ / 16) / NWAVE) // 16 k-tiles per wave
#define FLT_BIG 3.402823466e+38f

__global__ __launch_bounds__(256, 1)
void mcq_kernel(const float* __restrict__ x,
                const float* __restrict__ cb,
                const float* __restrict__ temp,
                const float* __restrict__ gum,
                float* __restrict__ out)
{
    float* sampleO = out;
    float* codeO   = out + (size_t)N_ * M_ * K_;
    float* oneHotO = codeO + (size_t)N_ * M_;
    float* logitO  = oneHotO + (size_t)N_ * M_ * K_;

    const int n0   = blockIdx.x * TN;
    const int m    = blockIdx.y;
    const int tid  = threadIdx.x;
    const int wave = tid >> 5;
    const int lane = tid & 31;
    const int lo   = lane & 15;   // column / row-within-tile selector
    const int hi   = lane >> 4;   // lane-half: selects K-pair (A/B) or M+8 (C/D)

    __shared__ float s_x2[TN];
    __shared__ float s_bLv[NWAVE][TN];
    __shared__ int   s_bLk[NWAVE][TN];
    __shared__ float s_bSv[NWAVE][TN];
    __shared__ int   s_bSk[NWAVE][TN];
    __shared__ int   s_codek[TN];
    __shared__ int   s_sampk[TN];

    // ---- Load A tile (16 rows x D=32) in WMMA 32-bit A layout -------------
    // For WMMA step s (d-base = 4*s): lane holds d = 4*s + hi*2 + {0,1}.
    v2f a[8];
    const float* xrow = x + (size_t)(n0 + lo) * (M_ * D_) + m * D_ + hi * 2;
    float x2p = 0.f;
#pragma unroll
    for (int s = 0; s < 8; ++s) {
        a[s] = *(const v2f*)(xrow + 4 * s);
        x2p += a[s].x * a[s].x + a[s].y * a[s].y;
    }
    // two lane-halves hold complementary D-halves of the same row
    float x2 = x2p + __shfl_xor(x2p, 16, 32);
    if (tid < TN) s_x2[tid] = x2;

    const float t    = temp[m];
    const float coef = -fmaxf(t, 1e-6f) * 0.022097086912079608f; // 1/sqrt(2048)

    float bLv[8], bSv[8];
    int   bLk[8], bSk[8];
#pragma unroll
    for (int r = 0; r < 8; ++r) {
        bLv[r] = -FLT_BIG; bSv[r] = -FLT_BIG; bLk[r] = 0; bSk[r] = 0;
    }

    __syncthreads();

    float rx2[8];
#pragma unroll
    for (int r = 0; r < 8; ++r) rx2[r] = s_x2[r + hi * 8];

    // ---- Main GEMM + elementwise streaming pass ---------------------------
    for (int t2 = 0; t2 < KT_PER_WAVE; ++t2) {
        const int k0 = (wave * KT_PER_WAVE + t2) * 16;
        const int kl = k0 + lo;
        // B (4x16 per step) mirrors A layout: lane-half picks K-pair,
        // component picks K within pair; column = lo -> codebook row k0+lo.
        // Codebook is small (1 MB) and reused by all workgroups: keep RT
        // (L2-resident) temporal hint for these loads.
        const float* crow = cb + ((size_t)m * K_ + kl) * D_ + hi * 2;
        v2f b[8];
        float c2p = 0.f;
#pragma unroll
        for (int s = 0; s < 8; ++s) {
            b[s] = *(const v2f*)(crow + 4 * s);
            c2p += b[s].x * b[s].x + b[s].y * b[s].y;
        }
        float c2 = c2p + __shfl_xor(c2p, 16, 32); // ||c_k||^2 for k = kl

        v8f acc = {0.f, 0.f, 0.f, 0.f, 0.f, 0.f, 0.f, 0.f};
#pragma unroll
        for (int s = 0; s < 8; ++s) {
            acc = __builtin_amdgcn_wmma_f32_16x16x4_f32(
                false, a[s], false, b[s], (short)0, acc, false, false);
        }

        // C/D layout: VGPR r -> row (r + hi*8), column lo -> k = kl
#pragma unroll
        for (int r = 0; r < 8; ++r) {
            const int    n   = n0 + r + hi * 8;
            const size_t idx = ((size_t)n * M_ + m) * K_ + kl;
            const float  lg  = (rx2[r] + c2 - 2.f * acc[r]) * coef;
            // gumbel is a 128 MB single-use stream: non-temporal load
            const float  g   = __builtin_nontemporal_load(gum + idx);
            // logit is a 128 MB never-re-read stream: non-temporal store
            __builtin_nontemporal_store(lg, logitO + idx);
            const float lgg = lg + g;
            if (lg  > bLv[r]) { bLv[r] = lg;  bLk[r] = kl; }
            if (lgg > bSv[r]) { bSv[r] = lgg; bSk[r] = kl; }
        }
    }

    // ---- Wave-level argmax reduce (within each lane-half of 16 lanes) -----
#pragma unroll
    for (int r = 0; r < 8; ++r) {
#pragma unroll
        for (int msk = 8; msk >= 1; msk >>= 1) {
            float ov = __shfl_xor(bLv[r], msk, 32);
            int   ok = __shfl_xor(bLk[r], msk, 32);
            if (ov > bLv[r] || (ov == bLv[r] && ok < bLk[r])) { bLv[r] = ov; bLk[r] = ok; }
            ov = __shfl_xor(bSv[r], msk, 32);
            ok = __shfl_xor(bSk[r], msk, 32);
            if (ov > bSv[r] || (ov == bSv[r] && ok < bSk[r])) { bSv[r] = ov; bSk[r] = ok; }
        }
        if (lo == 0) {
            s_bLv[wave][r + hi * 8] = bLv[r];
            s_bLk[wave][r + hi * 8] = bLk[r];
            s_bSv[wave][r + hi * 8] = bSv[r];
            s_bSk[wave][r + hi * 8] = bSk[r];
        }
    }
    __syncthreads();

    // ---- Cross-wave reduce; waves own ascending K chunks (tie -> min k) ---
    if (tid < TN) {
        float bv = -FLT_BIG, sv = -FLT_BIG;
        int   bk = 0, sk = 0;
        for (int w = 0; w < NWAVE; ++w) {
            float v = s_bLv[w][tid]; int kk = s_bLk[w][tid];
            if (v > bv || (v == bv && kk < bk)) { bv = v; bk = kk; }
            v = s_bSv[w][tid]; kk = s_bSk[w][tid];
            if (v > sv || (v == sv && kk < sk)) { sv = v; sk = kk; }
        }
        s_codek[tid] = bk;
        s_sampk[tid] = sk;
        codeO[(size_t)(n0 + tid) * M_ + m] = (float)bk;
    }
    __syncthreads();

    // ---- Phase 2: write one-hot sample & oneHot from broadcast indices ----
    // 256 MB of never-re-read output: non-temporal vector stores.
#pragma unroll 1
    for (int row = 0; row < TN; ++row) {
        const int    bs   = s_sampk[row];
        const int    bl   = s_codek[row];
        const size_t base = ((size_t)(n0 + row) * M_ + m) * K_;
        for (int j = tid; j < K_ / 4; j += 256) {
            const int k4 = j * 4;
            v4f vs, vo;
#pragma unroll
            for (int c = 0; c < 4; ++c) {
                vs[c] = (k4 + c == bs) ? 1.f : 0.f;
                vo[c] = (k4 + c == bl) ? 1.f : 0.f;
            }
            __builtin_nontemporal_store(vs, (v4f*)(sampleO + base + k4));
            __builtin_nontemporal_store(vo, (v4f*)(oneHotO + base + k4));
        }
    }
}

extern "C" void kernel_launch(void* const* d_in, const int* in_sizes, int n_in,
                              void* d_out, int out_size, void* d_ws, size_t ws_size,
                              hipStream_t stream) {
    (void)in_sizes; (void)n_in; (void)d_ws; (void)ws_size; (void)out_size;
    const float* x  = (const float*)d_in[0];
    const float* cb = (const float*)d_in[1];
    const float* tp = (const float*)d_in[2];
    const float* gm = (const float*)d_in[3];
    float* out = (float*)d_out;
    dim3 grid(N_ / TN, M_);
    dim3 block(256);
    hipLaunchKernelGGL(mcq_kernel, grid, block, 0, stream, x, cb, tp, gm, out);
}